// SpectralConvLocalSimp_66700842107121
// MI455X (gfx1250) — compile-verified
//
#include <hip/hip_runtime.h>
#include <hip/hip_bf16.h>

typedef __attribute__((ext_vector_type(2))) float v2f;
typedef __attribute__((ext_vector_type(8))) float v8f;

#define B_  4
#define N_  20000
#define E_  320000
#define CIN_ 32
#define COUT_ 32
#define K_  16
#define M_  2

// ---------------- zero workspace ----------------
__global__ __launch_bounds__(256) void zero_kernel(float* __restrict__ p, long n) {
    long i = (long)blockIdx.x * blockDim.x + threadIdx.x;
    long stride = (long)gridDim.x * blockDim.x;
    for (; i < n; i += stride) p[i] = 0.0f;
}

// ---------------- edge kernel: per-edge spectral weights via f32 WMMA ----------------
// One wave handles 16 edges for one (b, m).
// GEMM: A(16 edges x 32) [pc | ps]  x  B(32 x 32 cin) [2*wc ; 2*ws] -> edge_w tile.
// Done as 8 chunks of V_WMMA_F32_16X16X4_F32 per 16-cin half.
__global__ __launch_bounds__(256) void edge_kernel(
    const float* __restrict__ f,        // (B,CIN,N)
    const float* __restrict__ bases_c,  // (B,N,K,M)
    const float* __restrict__ bases_s,  // (B,N,K,M)
    const float* __restrict__ nodes,    // (B,N,2)
    const float* __restrict__ nvec,     // (B,N,2)
    const int*   __restrict__ edges,    // (B,E,2,M)
    const float* __restrict__ node_w,   // (B,E,M)
    const float* __restrict__ wc,       // (CIN,K,M)
    const float* __restrict__ ws,       // (CIN,K,M)
    const float* __restrict__ w0,       // (CIN,1,M)
    float* __restrict__ fout)           // (B,N,CIN) accumulator
{
    const int wave = (blockIdx.x * blockDim.x + threadIdx.x) >> 5;
    const int lane = threadIdx.x & 31;
    const int tilesPerBM = E_ / 16;               // 20000
    // grid sized exactly: no partial waves (EXEC must be all-ones for WMMA)
    const int tile = wave % tilesPerBM;
    const int bm   = wave / tilesPerBM;
    const int m = bm % M_;
    const int b = bm / M_;

    const int row = lane & 15;   // edge row within tile (duplicated across halves)
    const int hi  = lane >> 4;   // which K-pair half this lane supplies
    const int col = row;         // B/D column index (cin within half)

    // ---- per-edge scalars (each lane computes for its row; lanes 16-31 duplicate) ----
    const int  eidx  = tile * 16 + row;
    const long ebase = (long)b * E_ + eidx;
    const int  tgt = edges[(ebase * 2 + 0) * M_ + m];
    const int  src = edges[(ebase * 2 + 1) * M_ + m];

    const long tb2 = ((long)b * N_ + tgt) * 2;
    const long sb2 = ((long)b * N_ + src) * 2;
    const float dx = nodes[tb2 + 0] - nodes[sb2 + 0];
    const float dy = nodes[tb2 + 1] - nodes[sb2 + 1];
    const float nx = nvec[sb2 + 0];
    const float ny = nvec[sb2 + 1];
    const float r2 = dx * dx + dy * dy + 1e-6f;
    const float gradn = (dx * nx + dy * ny) / r2;
    const float g = node_w[ebase * M_ + m] * gradn;

    // ---- A-matrix features: lane supplies K-pairs {4q+2hi, 4q+2hi+1}, q=0..3 ----
    float apc[8], aps[8];
    const long tb = ((long)b * N_ + tgt) * K_;
    const long sb = ((long)b * N_ + src) * K_;
#pragma unroll
    for (int q = 0; q < 4; ++q) {
#pragma unroll
        for (int j = 0; j < 2; ++j) {
            const int k = 4 * q + 2 * hi + j;
            const float bct = bases_c[(tb + k) * M_ + m];
            const float bcs = bases_c[(sb + k) * M_ + m];
            const float bst = bases_s[(tb + k) * M_ + m];
            const float bss = bases_s[(sb + k) * M_ + m];
            apc[2 * q + j] = bct * bcs + bst * bss;   // cos-part product
            aps[2 * q + j] = bct * bss - bst * bcs;   // sin-part product
        }
    }

    // ---- B-matrix (weights): bw[h][kc][j], kc 0..3 from wc, 4..7 from ws ----
    float bw[2][8][2];
#pragma unroll
    for (int h = 0; h < 2; ++h) {
        const int i = col + 16 * h;
#pragma unroll
        for (int q = 0; q < 4; ++q) {
#pragma unroll
            for (int j = 0; j < 2; ++j) {
                const int k = 4 * q + 2 * hi + j;
                bw[h][q][j]     = 2.0f * wc[(i * K_ + k) * M_ + m];
                bw[h][q + 4][j] = 2.0f * ws[(i * K_ + k) * M_ + m];
            }
        }
    }

    // ---- 16 x WMMA f32 16x16x4, K=32 combined ----
    v8f c0 = {0.f, 0.f, 0.f, 0.f, 0.f, 0.f, 0.f, 0.f};
    v8f c1 = {0.f, 0.f, 0.f, 0.f, 0.f, 0.f, 0.f, 0.f};
#pragma unroll
    for (int kc = 0; kc < 8; ++kc) {
        v2f a;
        if (kc < 4) { a.x = apc[2 * kc + 0];      a.y = apc[2 * kc + 1]; }
        else        { a.x = aps[2 * (kc - 4) + 0]; a.y = aps[2 * (kc - 4) + 1]; }
        v2f b0v; b0v.x = bw[0][kc][0]; b0v.y = bw[0][kc][1];
        v2f b1v; b1v.x = bw[1][kc][0]; b1v.y = bw[1][kc][1];
        c0 = __builtin_amdgcn_wmma_f32_16x16x4_f32(false, a, false, b0v, (short)0, c0, false, false);
        c1 = __builtin_amdgcn_wmma_f32_16x16x4_f32(false, a, false, b1v, (short)0, c1, false, false);
    }

    // ---- apply per-edge scalar, gather f_s, scatter-add into f_out ----
    // D layout: lane (col,hi), VGPR j -> edge row = j + 8*hi, cin = col + 16*h
#pragma unroll
    for (int h = 0; h < 2; ++h) {
        const int cin = col + 16 * h;
        const float w0v = w0[cin * M_ + m];
        const v8f c = h ? c1 : c0;
#pragma unroll
        for (int j = 0; j < 8; ++j) {
            const int e = j + 8 * hi;
            const int   tgt_e = __shfl(tgt, e, 32);
            const int   src_e = __shfl(src, e, 32);
            const float g_e   = __shfl(g,   e, 32);
            const float fsv = f[((long)b * CIN_ + cin) * N_ + src_e];
            const float contrib = (c[j] + w0v + 1.0f) * fsv * g_e;
            atomicAdd(&fout[((long)b * N_ + tgt_e) * CIN_ + cin], contrib);
        }
    }
}

// ---------------- output channel mix: out = w_kernel * f_out^T + bias, via f32 WMMA ----------------
// One wave per 16-node tile per batch. A = w_kernel half (16 x 32), B = f_out^T (32 x 16 nodes).
__global__ __launch_bounds__(256) void out_kernel(
    const float* __restrict__ fout,  // (B,N,CIN)
    const float* __restrict__ wk,    // (COUT,CIN)
    const float* __restrict__ bias,  // (COUT,)
    float* __restrict__ out)         // (B,COUT,N)
{
    const int wave = (blockIdx.x * blockDim.x + threadIdx.x) >> 5;
    const int lane = threadIdx.x & 31;
    const int tilesPerB = N_ / 16;   // 1250
    const int b  = wave / tilesPerB;
    const int nt = wave % tilesPerB;
    const int col = lane & 15;
    const int hi  = lane >> 4;
    const int n = nt * 16 + col;

    v8f c0 = {0.f, 0.f, 0.f, 0.f, 0.f, 0.f, 0.f, 0.f};
    v8f c1 = {0.f, 0.f, 0.f, 0.f, 0.f, 0.f, 0.f, 0.f};
    const long fb = ((long)b * N_ + n) * CIN_;
#pragma unroll
    for (int kc = 0; kc < 8; ++kc) {
        const int k = 4 * kc + 2 * hi;
        v2f bv; bv.x = fout[fb + k];                 bv.y = fout[fb + k + 1];
        v2f a0; a0.x = wk[col * CIN_ + k];           a0.y = wk[col * CIN_ + k + 1];
        v2f a1; a1.x = wk[(col + 16) * CIN_ + k];    a1.y = wk[(col + 16) * CIN_ + k + 1];
        c0 = __builtin_amdgcn_wmma_f32_16x16x4_f32(false, a0, false, bv, (short)0, c0, false, false);
        c1 = __builtin_amdgcn_wmma_f32_16x16x4_f32(false, a1, false, bv, (short)0, c1, false, false);
    }

#pragma unroll
    for (int h = 0; h < 2; ++h) {
        const v8f c = h ? c1 : c0;
#pragma unroll
        for (int j = 0; j < 8; ++j) {
            const int o = j + 8 * hi + 16 * h;
            out[((long)b * COUT_ + o) * N_ + n] = c[j] + bias[o];
        }
    }
}

extern "C" void kernel_launch(void* const* d_in, const int* in_sizes, int n_in,
                              void* d_out, int out_size, void* d_ws, size_t ws_size,
                              hipStream_t stream) {
    const float* f        = (const float*)d_in[0];
    const float* bases_c  = (const float*)d_in[1];
    const float* bases_s  = (const float*)d_in[2];
    // d_in[3] = bases_0 : unused by the reference computation
    const float* nodes    = (const float*)d_in[4];
    const float* nvec     = (const float*)d_in[5];
    const int*   edges    = (const int*)d_in[6];
    const float* node_w   = (const float*)d_in[7];
    const float* wc       = (const float*)d_in[8];
    const float* ws       = (const float*)d_in[9];
    const float* w0       = (const float*)d_in[10];
    const float* wk       = (const float*)d_in[11];
    const float* bias     = (const float*)d_in[12];
    float* out = (float*)d_out;

    float* fout = (float*)d_ws;                       // (B,N,CIN) fp32 accumulator
    const long foutN = (long)B_ * N_ * CIN_;          // 2.56M floats = 10.24 MB

    zero_kernel<<<512, 256, 0, stream>>>(fout, foutN);

    // edge tiles: B * M * (E/16) = 160000 waves; 8 waves/block -> 20000 blocks (exact)
    const int edgeWaves = B_ * M_ * (E_ / 16);
    edge_kernel<<<edgeWaves / 8, 256, 0, stream>>>(
        f, bases_c, bases_s, nodes, nvec, edges, node_w, wc, ws, w0, fout);

    // out tiles: B * (N/16) = 5000 waves; 8 waves/block -> 625 blocks (exact)
    const int outWaves = B_ * (N_ / 16);
    out_kernel<<<outWaves / 8, 256, 0, stream>>>(fout, wk, bias, out);
}